// SegSPGSemLeaky_52974126629692
// MI455X (gfx1250) — compile-verified
//
#include <hip/hip_runtime.h>
#include <stdint.h>

namespace {
constexpr int   kC         = 32;
constexpr int   kNClust    = 2048;
constexpr int   kPts       = 1024;
constexpr int   kSumN      = kNClust * kPts;
constexpr float kFullScale = 14.0f;
constexpr float kScaleMax  = 50.0f;
constexpr int   kOutW      = kC + 3;   // 35 floats per output row
constexpr int   kBatch     = 4;        // rows per wave batch (one async b128 load)
constexpr int   kEmitBlocks = 4096;    // 4096*256/32 = 32768 waves; 2,097,152/(32768*4) = 16 batches/wave
}

// ---------------------------------------------------------------------------
// Phase 1: per-cluster raw min/max -> folded (scale, shift) parameters.
// Mean-centering cancels out of every downstream quantity (min/max/ext/rng are
// translation-invariant), so raw min/max of gathered coords is sufficient.
// ---------------------------------------------------------------------------
__global__ __launch_bounds__(256) void spg_cluster_stats(
    const float* __restrict__ coords,   // [N,3]
    const int*   __restrict__ cidx,     // [SUMN,2] = (cluster_id, point_id)
    const float* __restrict__ rand_a,   // [3]
    const float* __restrict__ rand_b,   // [3]
    float4*      __restrict__ params)   // [NCLUST] = (scale, shift.xyz)
{
  const int cluster = blockIdx.x;
  const int tid     = threadIdx.x;

  float mn[3] = {  3.402823466e38f,  3.402823466e38f,  3.402823466e38f };
  float mx[3] = { -3.402823466e38f, -3.402823466e38f, -3.402823466e38f };

  // 1024 points / 256 threads = 4 points per thread. Issue all 4 pid loads,
  // then all 4 coord gathers, to keep 4 loads in flight (MLP) instead of a
  // serialized pid->coords->pid chain.
  const int base = cluster * kPts + tid;
  int pids[4];
#pragma unroll
  for (int u = 0; u < 4; ++u) pids[u] = cidx[2 * (base + u * 256) + 1];
#pragma unroll
  for (int u = 0; u < 4; ++u) {
#pragma unroll
    for (int d = 0; d < 3; ++d) {
      const float v = coords[3 * pids[u] + d];
      mn[d] = fminf(mn[d], v);
      mx[d] = fmaxf(mx[d], v);
    }
  }

  // wave32 butterfly reduction
#pragma unroll
  for (int off = 16; off > 0; off >>= 1) {
#pragma unroll
    for (int d = 0; d < 3; ++d) {
      mn[d] = fminf(mn[d], __shfl_xor(mn[d], off, 32));
      mx[d] = fmaxf(mx[d], __shfl_xor(mx[d], off, 32));
    }
  }

  __shared__ float smn[3][8];
  __shared__ float smx[3][8];
  const int wave = tid >> 5;
  if ((tid & 31) == 0) {
#pragma unroll
    for (int d = 0; d < 3; ++d) { smn[d][wave] = mn[d]; smx[d][wave] = mx[d]; }
  }
  __syncthreads();

  if (tid == 0) {
    float fmn[3], fmx[3];
#pragma unroll
    for (int d = 0; d < 3; ++d) { fmn[d] = smn[d][0]; fmx[d] = smx[d][0]; }
    for (int w = 1; w < 8; ++w) {
#pragma unroll
      for (int d = 0; d < 3; ++d) {
        fmn[d] = fminf(fmn[d], smn[d][w]);
        fmx[d] = fmaxf(fmx[d], smx[d][w]);
      }
    }
    const float extm = fmaxf(fmx[0] - fmn[0], fmaxf(fmx[1] - fmn[1], fmx[2] - fmn[2]));
    const float scale = fminf(1.0f / (extm / kFullScale) - 0.01f, kScaleMax);
    float sh[3];
#pragma unroll
    for (int d = 0; d < 3; ++d) {
      const float rng = (fmx[d] - fmn[d]) * scale;
      sh[d] = -fmn[d] * scale
            + fmaxf(kFullScale - rng - 0.001f, 0.0f) * rand_a[d]
            + fminf(kFullScale - rng + 0.001f, 0.0f) * rand_b[d];
    }
    params[cluster] = make_float4(scale, sh[0], sh[1], sh[2]);
  }
}

// ---------------------------------------------------------------------------
// Phase 2: emit [SUMN, 35] rows, 4 rows per wave per batch.
//
//  - One global_load_async_to_lds_b128 gathers all 4 feature rows (512 B):
//    lane i fetches quad (i&7) of row (i>>3); feats rows are 128 B so every
//    per-lane source address is 16 B aligned.
//  - Lanes 0..11 transform the 12 coords (k=lane/3, d=lane%3) via VALU+store,
//    overlapping the async load.
//  - s_wait_asynccnt 0, then 4 global_store_async_from_lds_b32 (output rows
//    are 140 B apart -> only 4 B aligned, so B32 on the store side).
//  - Double-buffered LDS groups: the previous batch's outstanding async
//    stores read the other group, so this batch's load can be issued before
//    the wait. ASYNCcnt retires loads vs stores out of order, so the single
//    wait-0 per batch is the safe synchronization point.
// ---------------------------------------------------------------------------
__global__ __launch_bounds__(256) void spg_emit_async(
    const float*  __restrict__ feats,   // [N,32]
    const float*  __restrict__ coords,  // [N,3]
    const int*    __restrict__ cidx,    // [SUMN,2]
    const float4* __restrict__ params,  // [NCLUST]
    float*        __restrict__ out)     // [SUMN,35]
{
  __shared__ __attribute__((aligned(16))) float lbuf[8][2][kBatch][kC];

  const int lane   = threadIdx.x & 31;
  const int wv     = threadIdx.x >> 5;
  const int gwave  = (int)((blockIdx.x * blockDim.x + threadIdx.x) >> 5);
  const int nwaves = (int)((gridDim.x * blockDim.x) >> 5);

  // Low 32 bits of a __shared__ flat address = LDS byte offset (ISA 10.2).
  const unsigned grp0 = (unsigned)(uintptr_t)&lbuf[wv][0][0][0];
  const unsigned grp1 = (unsigned)(uintptr_t)&lbuf[wv][1][0][0];
  // Per-lane byte offset for the b128 gather: row (lane>>3), quad (lane&7).
  const unsigned ld_off = (unsigned)(((lane >> 3) * kC + (lane & 7) * 4) * 4);

  int sel = 0;
  for (int base = gwave * kBatch; base < kSumN; base += nwaves * kBatch) {
    const unsigned grp = sel ? grp1 : grp0;

    // ---- 1) one async b128 load moves all 4 feature rows into LDS --------
    const int   myrow = base + (lane >> 3);
    const int   mypid = cidx[2 * myrow + 1];
    const float* src  = feats + (size_t)mypid * kC + (lane & 7) * 4;
    const unsigned lds_ld = grp + ld_off;
    asm volatile("global_load_async_to_lds_b128 %0, %1, off"
                 :: "v"(lds_ld), "v"(src)
                 : "memory");

    // ---- 2) coords for the 4 rows on lanes 0..11 (overlaps the load) -----
    if (lane < 12) {
      const int k   = lane / 3;
      const int d   = lane - 3 * k;
      const int row = base + k;
      const int2 cp = ((const int2*)cidx)[row];   // (cluster_id, point_id)
      const float4 p = params[cp.x];
      const float shift = (d == 0) ? p.y : ((d == 1) ? p.z : p.w);
      const float c = coords[3 * cp.y + d];
      out[(size_t)row * kOutW + d] = fmaf(c, p.x, shift);
    }

    // ---- 3) drain: this batch's load + previous batch's stores -----------
    asm volatile("s_wait_asynccnt 0x0" ::: "memory");

    // ---- 4) stream the 4 rows from LDS to out (B32 per lane) -------------
#pragma unroll
    for (int k = 0; k < kBatch; ++k) {
      const int row = base + k;
      const float* dst = out + (size_t)row * kOutW + 3 + lane;
      const unsigned lds_st = grp + (unsigned)((k * kC + lane) * 4);
      asm volatile("global_store_async_from_lds_b32 %0, %1, off"
                   :: "v"(dst), "v"(lds_st)
                   : "memory");
    }
    sel ^= 1;
  }
  // s_endpgm performs an implicit wait-idle, draining the final async stores.
}

// ---------------------------------------------------------------------------
extern "C" void kernel_launch(void* const* d_in, const int* in_sizes, int n_in,
                              void* d_out, int out_size, void* d_ws, size_t ws_size,
                              hipStream_t stream) {
  const float* feats  = (const float*)d_in[0];
  const float* coords = (const float*)d_in[1];
  const int*   cidx   = (const int*)d_in[2];
  // d_in[3] = clusters_offset: unused, cluster size is fixed at 1024.
  const float* rand_a = (const float*)d_in[4];
  const float* rand_b = (const float*)d_in[5];

  float4* params = (float4*)d_ws;            // 2048 * 16 B = 32 KB scratch
  float*  out    = (float*)d_out;            // [SUMN, 35] float32

  spg_cluster_stats<<<kNClust, 256, 0, stream>>>(coords, cidx, rand_a, rand_b, params);
  spg_emit_async<<<kEmitBlocks, 256, 0, stream>>>(feats, coords, cidx, params, out);
}